// PointsPillarFeatureNetWrapper_48893907697728
// MI455X (gfx1250) — compile-verified
//
#include <hip/hip_runtime.h>

// ---- CDNA5 (gfx1250) wave32 WMMA types ----
typedef __attribute__((ext_vector_type(2))) float v2f;
typedef __attribute__((ext_vector_type(8))) float v8f;

#define VPIL 80000
#define PPTS 20
#define COUT 64
#define NXD  640
#define HW   (640 * 640)

constexpr float kVX   = 100.0f / 640.0f;              // 0.15625
constexpr float kXOFF = (100.0f / 640.0f) * 0.5f - 50.0f; // -49.921875
constexpr float kEPS  = 0.001f;

// ---------------------------------------------------------------------------
// Kernel 1: zero-fill the 426 MB output (canvas + occ). Pure store bandwidth.
// ---------------------------------------------------------------------------
__global__ __launch_bounds__(256) void fill_zero_kernel(float4* __restrict__ out4,
                                                        long long n4,
                                                        float* __restrict__ out,
                                                        long long n) {
  long long idx    = (long long)blockIdx.x * blockDim.x + threadIdx.x;
  long long stride = (long long)gridDim.x * blockDim.x;
  for (long long i = idx; i < n4; i += stride)
    out4[i] = make_float4(0.f, 0.f, 0.f, 0.f);
  long long t = 4 * n4 + idx;   // scalar tail (n is 4-divisible here, but be safe)
  if (t < n) out[t] = 0.f;
}

// ---------------------------------------------------------------------------
// Kernel 2: one wave32 per pillar. 128 threads = 4 waves = 4 pillars / block.
// Feature build -> LDS stage -> V_WMMA_F32_16X16X4_F32 GEMM -> BN+ReLU ->
// max over points -> scatter into NCHW canvas + occ.
// ---------------------------------------------------------------------------
__global__ __launch_bounds__(128) void pillar_kernel(
    const float* __restrict__ voxels,     // (V, P, 4)
    const int*   __restrict__ num_points, // (V)
    const int*   __restrict__ coors,      // (V, 4)
    const float* __restrict__ W,          // (9, 64)
    const float* __restrict__ gamma,
    const float* __restrict__ beta,
    const float* __restrict__ rmean,
    const float* __restrict__ rvar,
    float* __restrict__ out) {
  __shared__ float sW[12][COUT];     // W padded K 9->12 (zeros)
  __shared__ float sS[COUT];         // BN scale  = gamma * rsqrt(var+eps)
  __shared__ float sB[COUT];         // BN bias   = beta - mean*scale
  __shared__ float sF[4][32][12];    // per-wave padded 32x12 feature tile

  const int tid = threadIdx.x;

  // Stage padded W
  for (int i = tid; i < 12 * COUT; i += 128) {
    const int r = i >> 6, c = i & 63;
    sW[r][c] = (r < 9) ? W[r * COUT + c] : 0.0f;
  }
  // Stage folded BN params
  if (tid < COUT) {
    const float s = gamma[tid] * rsqrtf(rvar[tid] + kEPS);
    sS[tid] = s;
    sB[tid] = beta[tid] - rmean[tid] * s;
  }

  const int wave = tid >> 5;
  const int lane = tid & 31;
  const int v    = blockIdx.x * 4 + wave;
  const bool inb = (v < VPIL);   // wave-uniform

  int   npts = 1, bIdx = 0, yy = 0, xx = 0;
  float px = 0.f, py = 0.f, pz = 0.f, pi = 0.f;
  if (inb) {
    npts = num_points[v];
    if (npts < 1) npts = 1;
    bIdx = coors[4 * v + 0];
    yy   = coors[4 * v + 2];
    xx   = coors[4 * v + 3];
    if (lane < PPTS) {
      const float* vp = voxels + ((long long)v * PPTS + lane) * 4;
      px = vp[0]; py = vp[1]; pz = vp[2]; pi = vp[3];
    }
  }
  const bool valid = inb && (lane < PPTS) && (lane < npts);

  // Cluster mean over valid points (wave32 reduction)
  float sx = valid ? px : 0.f, sy = valid ? py : 0.f, sz = valid ? pz : 0.f;
#pragma unroll
  for (int o = 16; o > 0; o >>= 1) {
    sx += __shfl_xor(sx, o);
    sy += __shfl_xor(sy, o);
    sz += __shfl_xor(sz, o);
  }
  const float inv = 1.0f / (float)npts;
  const float mx = sx * inv, my = sy * inv, mz = sz * inv;
  const float cx = (float)xx * kVX + kXOFF;
  const float cy = (float)yy * kVX + kXOFF;

  // Build 9 features, zero invalid rows (matches reference feats*valid)
  const float vm = valid ? 1.0f : 0.0f;
  float* fr = &sF[wave][lane][0];
  fr[0]  = px * vm;
  fr[1]  = py * vm;
  fr[2]  = pz * vm;
  fr[3]  = pi * vm;
  fr[4]  = (px - mx) * vm;
  fr[5]  = (py - my) * vm;
  fr[6]  = (pz - mz) * vm;
  fr[7]  = (px - cx) * vm;
  fr[8]  = (py - cy) * vm;
  fr[9]  = 0.0f;
  fr[10] = 0.0f;
  fr[11] = 0.0f;

  __syncthreads();

  // ---- GEMM via V_WMMA_F32_16X16X4_F32: (32x12) x (12x64) ----
  const int  l16 = lane & 15;
  const bool hi  = lane >= 16;
  const int  kSh = hi ? 2 : 0;   // ISA A/B fragment layout: hi half holds K+2..K+3

  if (inb) {
#pragma unroll
    for (int nt = 0; nt < 4; ++nt) {
      const int n0 = nt * 16;
      v8f acc0 = {};   // rows (points) 0..15
      v8f acc1 = {};   // rows 16..31 (16..19 real, rest pad)
#pragma unroll
      for (int kc = 0; kc < 3; ++kc) {
        const int k0 = kc * 4 + kSh;
        v2f bfr; bfr.x = sW[k0][n0 + l16];      bfr.y = sW[k0 + 1][n0 + l16];
        v2f a0;  a0.x  = sF[wave][l16][k0];     a0.y  = sF[wave][l16][k0 + 1];
        v2f a1;  a1.x  = sF[wave][16 + l16][k0]; a1.y = sF[wave][16 + l16][k0 + 1];
        acc0 = __builtin_amdgcn_wmma_f32_16x16x4_f32(
            false, a0, false, bfr, (short)0, acc0, false, false);
        acc1 = __builtin_amdgcn_wmma_f32_16x16x4_f32(
            false, a1, false, bfr, (short)0, acc1, false, false);
      }

      // BN + ReLU + max over point rows. Channel of this lane: n0 + l16.
      const float s = sS[n0 + l16];
      const float t = sB[n0 + l16];
      float m0 = -3.0e38f;
#pragma unroll
      for (int i = 0; i < 8; ++i)                       // rows 0..15
        m0 = fmaxf(m0, fmaxf(acc0[i] * s + t, 0.0f));
      if (!hi) {
#pragma unroll
        for (int i = 0; i < 4; ++i)                     // rows 16..19 only
          m0 = fmaxf(m0, fmaxf(acc1[i] * s + t, 0.0f));
      }
      m0 = fmaxf(m0, __shfl_xor(m0, 16));               // combine row halves

      if (!hi) {
        // NCHW canvas scatter: out[b][c][y][x]
        out[((long long)bIdx * COUT + (n0 + l16)) * HW + (long long)yy * NXD + xx] = m0;
      }
    }
    if (lane == 0) {
      const long long occBase = (long long)4 * COUT * HW;
      out[occBase + (long long)bIdx * HW + (long long)yy * NXD + xx] = 1.0f;
    }
  }
}

// ---------------------------------------------------------------------------
extern "C" void kernel_launch(void* const* d_in, const int* in_sizes, int n_in,
                              void* d_out, int out_size, void* d_ws, size_t ws_size,
                              hipStream_t stream) {
  const float* voxels     = (const float*)d_in[0];
  const int*   num_points = (const int*)d_in[1];
  const int*   coors      = (const int*)d_in[2];
  const float* W          = (const float*)d_in[3];
  const float* gamma      = (const float*)d_in[4];
  const float* beta       = (const float*)d_in[5];
  const float* rmean      = (const float*)d_in[6];
  const float* rvar       = (const float*)d_in[7];
  float* out = (float*)d_out;

  const long long n  = (long long)out_size;
  const long long n4 = n >> 2;
  long long fbll = (n4 + 255) / 256;
  int fb = (fbll > 131072) ? 131072 : (int)fbll;   // grid-stride covers the rest
  if (fb < 1) fb = 1;
  fill_zero_kernel<<<fb, 256, 0, stream>>>((float4*)out, n4, out, n);

  pillar_kernel<<<(VPIL + 3) / 4, 128, 0, stream>>>(
      voxels, num_points, coors, W, gamma, beta, rmean, rvar, out);
}